// DTIGCN_5626407158101
// MI455X (gfx1250) — compile-verified
//
#include <hip/hip_runtime.h>
#include <math.h>

// ---------------------------------------------------------------------------
// Problem dims (fixed by the reference)
#define ND  4000
#define NP  4000
#define NDI 3000
#define NSE 2000
#define DF  64

#define KSPLIT   8      // K-dimension splits per spmm (blockIdx.y)
#define PMAXROWS 4000   // partial buffer row capacity
#define PSTRIDE  68     // 64 cols + rowsum @ [64] + pad

typedef __attribute__((ext_vector_type(2))) float v2f;
typedef __attribute__((ext_vector_type(8))) float v8f;

typedef __attribute__((address_space(3))) void lds_void;

// LDS byte offset of a generic pointer into the block's LDS allocation
__device__ __forceinline__ unsigned ldsOffset(void* p) {
    return (unsigned)(unsigned long long)(lds_void*)p;
}

// CDNA5 async global -> LDS copy, 16B per active lane, tracked by ASYNCcnt
__device__ __forceinline__ void asyncLoad16(unsigned ldsoff, const void* g) {
    asm volatile("global_load_async_to_lds_b128 %0, %1, off"
                 :: "v"(ldsoff), "v"(g)
                 : "memory");
}

// ---------------------------------------------------------------------------
// Phase 1: HT = act(x @ W^T + b)^T  stored TRANSPOSED: HT[c][row], c=0..63
// act: 0 = relu, 1 = elu
template <int ACT>
__global__ void linear_act_kernel(const float* __restrict__ x,
                                  const float* __restrict__ W,
                                  const float* __restrict__ b,
                                  float* __restrict__ HT, int N) {
    __shared__ float sWt[64 * 65];   // sWt[d*65 + j] = W[j][d]
    __shared__ float sb[64];
    __shared__ float sx[4][64];

    const int t = threadIdx.x;       // 256 threads
    for (int i = t; i < 64 * 64; i += 256) {
        int j = i >> 6, d = i & 63;
        sWt[d * 65 + j] = W[i];
    }
    if (t < 64) sb[t] = b[t];

    const int r = t >> 6;            // 0..3
    const int c = t & 63;            // 0..63 (output col j)
    const int gr = blockIdx.x * 4 + r;
    sx[r][c] = (gr < N) ? x[(size_t)gr * DF + c] : 0.f;
    __syncthreads();

    if (gr < N) {
        float a = sb[c];
#pragma unroll
        for (int d = 0; d < 64; ++d)
            a = fmaf(sx[r][d], sWt[d * 65 + c], a);
        if (ACT == 0) a = a > 0.f ? a : 0.f;
        else          a = a > 0.f ? a : expm1f(a);
        HT[(size_t)c * N + gr] = a;   // transposed store
    }
}

// ---------------------------------------------------------------------------
// Phase 2a: per-K-split partials of A @ H and rowsum(A).
// A:(M,K) f32, HT:(64,K) f32.  128 threads = 4 waves; wave w owns rows
// [64*blkX+16w, +16) x 64 cols, K range [blkY*Klen, min(K, +Klen)).
// Double-buffered LDS tiles filled with global_load_async_to_lds_b128;
// rowsum fused as a 5th WMMA with an all-ones B fragment.
#define KC   64
#define LDA  68
#define LDHT 68

__device__ __forceinline__ void stage_chunk(float* sAb, float* sHb,
                                            const float* __restrict__ A,
                                            const float* __restrict__ HT,
                                            int rowBlock, int k0, int M, int K,
                                            int kend, int t) {
    // A tile: 64 rows x KC cols. Rows clamped (garbage rows dropped at
    // writeback), K-tail lanes zero-filled via plain LDS stores. Every slot
    // instruction always has >=1 active lane, so each wave issues exactly
    // 16 async ops per stage (the s_wait_asynccnt 0x10 invariant).
#pragma unroll
    for (int i = 0; i < 8; ++i) {
        int idx = t + 128 * i;              // 0..1023
        int r   = idx >> 4;                 // 0..63
        int c4  = (idx & 15) << 2;          // 0..60 step 4
        int gr  = rowBlock + r; if (gr > M - 1) gr = M - 1;
        int gc  = k0 + c4;
        float* slot = sAb + r * LDA + c4;
        if (gc < kend) asyncLoad16(ldsOffset(slot), A + (size_t)gr * K + gc);
        else           *(float4*)slot = make_float4(0.f, 0.f, 0.f, 0.f);
    }
    // HT tile: 64 feature-rows x KC k-cols (contiguous along K in memory)
#pragma unroll
    for (int i = 0; i < 8; ++i) {
        int idx = t + 128 * i;
        int hr  = idx >> 4;                 // feature col 0..63
        int c4  = (idx & 15) << 2;
        int gc  = k0 + c4;
        float* slot = sHb + hr * LDHT + c4;
        if (gc < kend) asyncLoad16(ldsOffset(slot), HT + (size_t)hr * K + gc);
        else           *(float4*)slot = make_float4(0.f, 0.f, 0.f, 0.f);
    }
}

__global__ void __launch_bounds__(128)
spmm_partial_kernel(const float* __restrict__ A, const float* __restrict__ HT,
                    float* __restrict__ P, int M, int K, int Klen) {
    __shared__ float sA[2][64 * LDA];
    __shared__ float sH[2][64 * LDHT];

    const int t    = threadIdx.x;      // 0..127
    const int wave = t >> 5;
    const int lane = t & 31;
    const int half = lane >> 4;
    const int l16  = lane & 15;
    const int rowBlock = blockIdx.x * 64;
    const int split    = blockIdx.y;
    const int kbeg = split * Klen;
    const int kend = (kbeg + Klen < K) ? (kbeg + Klen) : K;

    v8f accC[4];
    v8f accS;
#pragma unroll
    for (int i = 0; i < 8; ++i) {
        accS[i] = 0.f;
#pragma unroll
        for (int c = 0; c < 4; ++c) accC[c][i] = 0.f;
    }
    v2f ones; ones.x = 1.f; ones.y = 1.f;

    const int nCh = (kend > kbeg) ? (kend - kbeg + KC - 1) / KC : 0;
    if (nCh > 0) {
        stage_chunk(sA[0], sH[0], A, HT, rowBlock, kbeg, M, K, kend, t);

        for (int ch = 0; ch < nCh; ++ch) {
            const int cur = ch & 1;
            if (ch + 1 < nCh) {
                stage_chunk(sA[cur ^ 1], sH[cur ^ 1], A, HT, rowBlock,
                            kbeg + (ch + 1) * KC, M, K, kend, t);
                // 16 ops of the next stage may remain in flight; everything
                // older (the current stage) must be complete.
                asm volatile("s_wait_asynccnt 0x10" ::: "memory");
            } else {
                asm volatile("s_wait_asynccnt 0x0" ::: "memory");
            }
            __syncthreads();

            const float* aRow = &sA[cur][(wave * 16 + l16) * LDA];
            const float* h0   = &sH[cur][(0 * 16 + l16) * LDHT];
            const float* h1   = &sH[cur][(1 * 16 + l16) * LDHT];
            const float* h2   = &sH[cur][(2 * 16 + l16) * LDHT];
            const float* h3   = &sH[cur][(3 * 16 + l16) * LDHT];
#pragma unroll
            for (int kc = 0; kc < KC; kc += 4) {
                const int ko = kc + 2 * half;
                // A fragment: VGPR p <-> K = ko + p (contiguous -> ds b64)
                v2f af; af.x = aRow[ko]; af.y = aRow[ko + 1];
                v2f b0; b0.x = h0[ko]; b0.y = h0[ko + 1];
                v2f b1; b1.x = h1[ko]; b1.y = h1[ko + 1];
                v2f b2; b2.x = h2[ko]; b2.y = h2[ko + 1];
                v2f b3; b3.x = h3[ko]; b3.y = h3[ko + 1];
                accC[0] = __builtin_amdgcn_wmma_f32_16x16x4_f32(false, af, false, b0, (short)0, accC[0], false, false);
                accC[1] = __builtin_amdgcn_wmma_f32_16x16x4_f32(false, af, false, b1, (short)0, accC[1], false, false);
                accC[2] = __builtin_amdgcn_wmma_f32_16x16x4_f32(false, af, false, b2, (short)0, accC[2], false, false);
                accC[3] = __builtin_amdgcn_wmma_f32_16x16x4_f32(false, af, false, b3, (short)0, accC[3], false, false);
                accS    = __builtin_amdgcn_wmma_f32_16x16x4_f32(false, af, false, ones, (short)0, accS, false, false);
            }
            __syncthreads();
        }
    }

    // writeback raw partials: P[split][row][0..63] = C, P[split][row][64] = rowsum
#pragma unroll
    for (int p = 0; p < 8; ++p) {
        int row = rowBlock + wave * 16 + p + 8 * half;   // C: VGPR p <-> M = p+8*half
        if (row < M) {
            float* pr = P + ((size_t)split * PMAXROWS + row) * PSTRIDE;
#pragma unroll
            for (int c = 0; c < 4; ++c)
                pr[c * 16 + l16] = accC[c][p];
            if (l16 == 0) pr[64] = accS[p];
        }
    }
}

// ---------------------------------------------------------------------------
// Phase 2b: deterministic cross-split reduction + row-normalize + accumulate
__global__ void reduce_partials_kernel(const float* __restrict__ P,
                                       float* __restrict__ acc, int M) {
    int idx = blockIdx.x * blockDim.x + threadIdx.x;
    if (idx >= M * DF) return;
    int row = idx >> 6;
    int col = idx & 63;
    float s = 0.f, rs = 0.f;
#pragma unroll
    for (int i = 0; i < KSPLIT; ++i) {
        const float* p = P + ((size_t)i * PMAXROWS + row) * PSTRIDE;
        s  += p[col];
        rs += p[64];
    }
    acc[idx] += s / rs;
}

// ---------------------------------------------------------------------------
// Phase 3: out = l2norm(acc * invCount), eps-clamped
__global__ void finalize_kernel(const float* __restrict__ acc,
                                float* __restrict__ out, int N, float invCount) {
    int row = blockIdx.x * blockDim.x + threadIdx.x;
    if (row >= N) return;
    const float* a = acc + (size_t)row * DF;
    float v[DF], ss = 0.f;
#pragma unroll
    for (int i = 0; i < DF; ++i) { v[i] = a[i] * invCount; ss += v[i] * v[i]; }
    float inv = 1.f / fmaxf(sqrtf(ss), 1e-12f);
    float* o = out + (size_t)row * DF;
#pragma unroll
    for (int i = 0; i < DF; ++i) o[i] = v[i] * inv;
}

// ---------------------------------------------------------------------------
extern "C" void kernel_launch(void* const* d_in, const int* in_sizes, int n_in,
                              void* d_out, int out_size, void* d_ws, size_t ws_size,
                              hipStream_t stream) {
    const float* feat_drug       = (const float*)d_in[0];
    const float* feat_protein    = (const float*)d_in[1];
    const float* feat_disease    = (const float*)d_in[2];
    const float* feat_sideeffect = (const float*)d_in[3];
    const float* A_dd  = (const float*)d_in[4];
    const float* A_dp  = (const float*)d_in[5];
    const float* A_ddi = (const float*)d_in[6];
    const float* A_dse = (const float*)d_in[7];
    const float* A_pp  = (const float*)d_in[8];
    const float* A_pd  = (const float*)d_in[9];
    const float* A_pdi = (const float*)d_in[10];
    const float* A_did = (const float*)d_in[11];
    const float* A_dip = (const float*)d_in[12];
    const float* A_sed = (const float*)d_in[13];
    const float* W_dd  = (const float*)d_in[14]; const float* b_dd  = (const float*)d_in[15];
    const float* W_dp  = (const float*)d_in[16]; const float* b_dp  = (const float*)d_in[17];
    const float* W_ddi = (const float*)d_in[18]; const float* b_ddi = (const float*)d_in[19];
    const float* W_dse = (const float*)d_in[20]; const float* b_dse = (const float*)d_in[21];
    const float* W_pp  = (const float*)d_in[22]; const float* b_pp  = (const float*)d_in[23];
    const float* W_pd  = (const float*)d_in[24]; const float* b_pd  = (const float*)d_in[25];
    const float* W_pdi = (const float*)d_in[26]; const float* b_pdi = (const float*)d_in[27];
    const float* W_did = (const float*)d_in[28]; const float* b_did = (const float*)d_in[29];
    const float* W_dip = (const float*)d_in[30]; const float* b_dip = (const float*)d_in[31];
    const float* W_sed = (const float*)d_in[32]; const float* b_sed = (const float*)d_in[33];
    float* out = (float*)d_out;

    // Workspace layout (floats)
    float* ws        = (float*)d_ws;
    float* acc_drug  = ws;                                   // 4000*64
    float* acc_prot  = acc_drug + (size_t)ND  * DF;          // 4000*64
    float* acc_dis   = acc_prot + (size_t)NP  * DF;          // 3000*64
    float* acc_se    = acc_dis  + (size_t)NDI * DF;          // 2000*64
    float* Hbuf      = acc_se   + (size_t)NSE * DF;          // 64*4000 (HT)
    float* Pbuf      = Hbuf     + (size_t)DF  * ND;          // 8*4000*68 partials

    // Seed accumulators with self-features
    hipMemcpyAsync(acc_drug, feat_drug,       (size_t)ND  * DF * 4, hipMemcpyDeviceToDevice, stream);
    hipMemcpyAsync(acc_prot, feat_protein,    (size_t)NP  * DF * 4, hipMemcpyDeviceToDevice, stream);
    hipMemcpyAsync(acc_dis,  feat_disease,    (size_t)NDI * DF * 4, hipMemcpyDeviceToDevice, stream);
    hipMemcpyAsync(acc_se,   feat_sideeffect, (size_t)NSE * DF * 4, hipMemcpyDeviceToDevice, stream);

    auto run_rel = [&](const float* A, const float* x, const float* W, const float* b,
                       float* accDst, int M, int Ksrc, int act) {
        dim3 g1((Ksrc + 3) / 4);
        if (act == 0)
            linear_act_kernel<0><<<g1, 256, 0, stream>>>(x, W, b, Hbuf, Ksrc);
        else
            linear_act_kernel<1><<<g1, 256, 0, stream>>>(x, W, b, Hbuf, Ksrc);
        // K split length: multiple of KC, KSPLIT splits cover K (all non-empty
        // for the fixed problem sizes)
        int Klen = ((Ksrc + KSPLIT - 1) / KSPLIT + KC - 1) / KC * KC;
        dim3 g2((M + 63) / 64, KSPLIT);
        spmm_partial_kernel<<<g2, 128, 0, stream>>>(A, Hbuf, Pbuf, M, Ksrc, Klen);
        reduce_partials_kernel<<<(M * DF + 255) / 256, 256, 0, stream>>>(Pbuf, accDst, M);
    };

    // Drug messages (relu)
    run_rel(A_dd,  feat_drug,       W_dd,  b_dd,  acc_drug, ND, ND,  0);
    run_rel(A_dp,  feat_protein,    W_dp,  b_dp,  acc_drug, ND, NP,  0);
    run_rel(A_ddi, feat_disease,    W_ddi, b_ddi, acc_drug, ND, NDI, 0);
    run_rel(A_dse, feat_sideeffect, W_dse, b_dse, acc_drug, ND, NSE, 0);
    // Protein messages (elu)
    run_rel(A_pp,  feat_protein,    W_pp,  b_pp,  acc_prot, NP, NP,  1);
    run_rel(A_pd,  feat_drug,       W_pd,  b_pd,  acc_prot, NP, ND,  1);
    run_rel(A_pdi, feat_disease,    W_pdi, b_pdi, acc_prot, NP, NDI, 1);
    // Disease messages (elu)
    run_rel(A_did, feat_drug,       W_did, b_did, acc_dis,  NDI, ND, 1);
    run_rel(A_dip, feat_protein,    W_dip, b_dip, acc_dis,  NDI, NP, 1);
    // Side-effect message (elu)
    run_rel(A_sed, feat_drug,       W_sed, b_sed, acc_se,   NSE, ND, 1);

    // Finalize: mean + L2 norm. Output order: drug, protein, sideeffect, disease
    finalize_kernel<<<(ND  + 255) / 256, 256, 0, stream>>>(acc_drug, out,                                ND,  1.f / 5.f);
    finalize_kernel<<<(NP  + 255) / 256, 256, 0, stream>>>(acc_prot, out + (size_t)ND * DF,              NP,  1.f / 4.f);
    finalize_kernel<<<(NSE + 255) / 256, 256, 0, stream>>>(acc_se,   out + (size_t)(ND + NP) * DF,       NSE, 1.f / 2.f);
    finalize_kernel<<<(NDI + 255) / 256, 256, 0, stream>>>(acc_dis,  out + (size_t)(ND + NP + NSE) * DF, NDI, 1.f / 3.f);
}